// GCN_38611755991789
// MI455X (gfx1250) — compile-verified
//
#include <hip/hip_runtime.h>

typedef float v2f __attribute__((ext_vector_type(2)));
typedef float v8f __attribute__((ext_vector_type(8)));

// ------------------------------------------------------------------
// degree / normalization
// ------------------------------------------------------------------
__global__ void k_deg_init(float* __restrict__ deg, int n) {
    int i = blockIdx.x * blockDim.x + threadIdx.x;
    if (i < n) deg[i] = 1.0f;                      // self-loop contribution
}

__global__ void k_deg_accum(const long long* __restrict__ dst,
                            float* __restrict__ deg, int e) {
    int i = blockIdx.x * blockDim.x + threadIdx.x;
    if (i < e) atomicAdd(&deg[(int)dst[i]], 1.0f);
}

__global__ void k_rsqrt_inplace(float* __restrict__ deg, int n) {
    int i = blockIdx.x * blockDim.x + threadIdx.x;
    if (i < n) deg[i] = rsqrtf(deg[i]);            // deg >= 1 always
}

// ------------------------------------------------------------------
// H = X @ W  via V_WMMA_F32_16X16X4_F32.
// One wave computes a 16 x OUT_C strip. IN_C fixed at 128.
// W (IN_C x OUT_C, row major) staged in LDS once per block.
// A frag (16x4 f32, 2 VGPR): lanes 0-15 -> K=k,k+1 ; lanes 16-31 -> K=k+2,k+3
// B frag (4x16 f32, 2 VGPR): same half-wave K split, N = lane&15
// C/D (16x16 f32, 8 VGPR):  vgpr r -> rows r (lanes 0-15) and r+8 (lanes 16-31)
// ------------------------------------------------------------------
template <int OUT_C>
__global__ __launch_bounds__(256) void k_gemm_wmma(const float* __restrict__ X,
                                                   const float* __restrict__ W,
                                                   float* __restrict__ H,
                                                   int n_tiles) {
    constexpr int IN_C = 128;
    __shared__ float Wlds[IN_C * OUT_C];           // 64KB (OUT_C=128) / 32KB (64)

    const int tid = threadIdx.x;
    for (int i = tid * 4; i < IN_C * OUT_C; i += 256 * 4) {
        float4 w = *(const float4*)(W + i);
        *(float4*)(Wlds + i) = w;
    }
    __syncthreads();

    const int wave = tid >> 5;
    const int lane = tid & 31;
    const int tile = blockIdx.x * 8 + wave;
    if (tile >= n_tiles) return;                   // wave-uniform: EXEC stays all-1s

    const int  row  = tile * 16 + (lane & 15);
    const int  koff = (lane < 16) ? 0 : 2;
    const float* xrow = X + (size_t)row * IN_C + koff;

    constexpr int NT = OUT_C / 16;
    v8f acc[NT] = {};

    for (int k = 0; k < IN_C; k += 4) {
        v2f a;
        a.x = xrow[k];
        a.y = xrow[k + 1];
        const float* wb = &Wlds[(k + koff) * OUT_C + (lane & 15)];
#pragma unroll
        for (int t = 0; t < NT; ++t) {
            v2f b;
            b.x = wb[t * 16];                      // row k+koff
            b.y = wb[OUT_C + t * 16];              // row k+koff+1
            acc[t] = __builtin_amdgcn_wmma_f32_16x16x4_f32(
                false, a, false, b, (short)0, acc[t], false, false);
        }
    }

    const int col0  = lane & 15;
    const int rhalf = (lane < 16) ? 0 : 8;
    float* hrow = H + (size_t)tile * 16 * OUT_C;
#pragma unroll
    for (int t = 0; t < NT; ++t)
#pragma unroll
        for (int r = 0; r < 8; ++r)
            hrow[(size_t)(r + rhalf) * OUT_C + t * 16 + col0] = acc[t][r];
}

// ------------------------------------------------------------------
// OUT[i,f] = H[i,f] * dinv[i]^2 + b[f]   (self-loop edge + bias)
// ------------------------------------------------------------------
template <int C>
__global__ void k_init_out(const float* __restrict__ H,
                           const float* __restrict__ dinv,
                           const float* __restrict__ bias,
                           float* __restrict__ OUT, int n) {
    long long i = (long long)blockIdx.x * blockDim.x + threadIdx.x;
    long long total = (long long)n * C;
    if (i >= total) return;
    int node = (int)(i / C);
    int f    = (int)(i % C);
    float d  = dinv[node];
    OUT[i] = H[i] * d * d + bias[f];
}

// ------------------------------------------------------------------
// Edge scatter: OUT[dst] += H[src] * dinv[src]*dinv[dst]
// one thread per (edge, 4-feature group); gathers hit L2 (H fits in 192MB)
// ------------------------------------------------------------------
template <int C>
__global__ void k_scatter(const long long* __restrict__ src,
                          const long long* __restrict__ dst,
                          const float* __restrict__ dinv,
                          const float* __restrict__ H,
                          float* __restrict__ OUT, int e) {
    constexpr int G = C / 4;
    long long i = (long long)blockIdx.x * blockDim.x + threadIdx.x;
    long long total = (long long)e * G;
    if (i >= total) return;
    int ei = (int)(i / G);
    int g  = (int)(i % G) * 4;
    int s  = (int)src[ei];
    int d  = (int)dst[ei];
    float norm = dinv[s] * dinv[d];
    float4 h = *(const float4*)(H + (size_t)s * C + g);
    float* o = OUT + (size_t)d * C + g;
    atomicAdd(o + 0, h.x * norm);
    atomicAdd(o + 1, h.y * norm);
    atomicAdd(o + 2, h.z * norm);
    atomicAdd(o + 3, h.w * norm);
}

__global__ void k_relu(float* __restrict__ x, long long total) {
    long long i = (long long)blockIdx.x * blockDim.x + threadIdx.x;
    if (i < total) x[i] = fmaxf(x[i], 0.0f);
}

// ------------------------------------------------------------------
static inline int cdiv(long long a, long long b) { return (int)((a + b - 1) / b); }

extern "C" void kernel_launch(void* const* d_in, const int* in_sizes, int n_in,
                              void* d_out, int out_size, void* d_ws, size_t ws_size,
                              hipStream_t stream) {
    const float* x  = (const float*)d_in[0];
    const float* W1 = (const float*)d_in[1];
    const float* b1 = (const float*)d_in[2];
    const float* W2 = (const float*)d_in[3];
    const float* b2 = (const float*)d_in[4];
    const float* W3 = (const float*)d_in[5];
    const float* b3 = (const float*)d_in[6];
    const long long* edge = (const long long*)d_in[7];

    const int IN_C = 128, HID_C = 128, OUT_C = 64;
    const int n = in_sizes[0] / IN_C;        // 100000 (divisible by 16)
    const int e = in_sizes[7] / 2;           // 1600000
    const long long* e_src = edge;
    const long long* e_dst = edge + e;

    // workspace layout (256B-aligned regions)
    char* ws = (char*)d_ws;
    size_t degBytes = ((size_t)n * 4 + 255) & ~(size_t)255;
    size_t matBytes = ((size_t)n * 128 * 4 + 255) & ~(size_t)255;
    float* dinv = (float*)ws;                 // n floats (deg -> rsqrt in place)
    float* bufH = (float*)(ws + degBytes);    // n x 128
    float* bufA = (float*)(ws + degBytes + matBytes); // n x 128
    float* out  = (float*)d_out;              // n x 64

    const int T = 256;
    const int tiles = n / 16;                 // exact
    const int gemmBlocks = cdiv(tiles, 8);

    // normalization: deg = 1 + count(dst); dinv = rsqrt(deg)
    k_deg_init<<<cdiv(n, T), T, 0, stream>>>(dinv, n);
    k_deg_accum<<<cdiv(e, T), T, 0, stream>>>(e_dst, dinv, e);
    k_rsqrt_inplace<<<cdiv(n, T), T, 0, stream>>>(dinv, n);

    // ---- layer 1: x -> bufA ----
    k_gemm_wmma<128><<<gemmBlocks, T, 0, stream>>>(x, W1, bufH, tiles);
    k_init_out<128><<<cdiv((long long)n * 128, T), T, 0, stream>>>(bufH, dinv, b1, bufA, n);
    k_scatter<128><<<cdiv((long long)e * 32, T), T, 0, stream>>>(e_src, e_dst, dinv, bufH, bufA, e);
    k_relu<<<cdiv((long long)n * 128, T), T, 0, stream>>>(bufA, (long long)n * 128);

    // ---- layer 2: bufA -> bufA (via bufH) ----
    k_gemm_wmma<128><<<gemmBlocks, T, 0, stream>>>(bufA, W2, bufH, tiles);
    k_init_out<128><<<cdiv((long long)n * 128, T), T, 0, stream>>>(bufH, dinv, b2, bufA, n);
    k_scatter<128><<<cdiv((long long)e * 32, T), T, 0, stream>>>(e_src, e_dst, dinv, bufH, bufA, e);
    k_relu<<<cdiv((long long)n * 128, T), T, 0, stream>>>(bufA, (long long)n * 128);

    // ---- layer 3: bufA -> d_out (no relu) ----
    k_gemm_wmma<64><<<gemmBlocks, T, 0, stream>>>(bufA, W3, bufH, tiles);
    k_init_out<64><<<cdiv((long long)n * 64, T), T, 0, stream>>>(bufH, dinv, b3, out, n);
    k_scatter<64><<<cdiv((long long)e * 16, T), T, 0, stream>>>(e_src, e_dst, dinv, bufH, out, e);
}